// StructureAttention_57226144252785
// MI455X (gfx1250) — compile-verified
//
#include <hip/hip_runtime.h>

// ---------------- problem constants ----------------
constexpr int Nn = 4;
constexpr int L  = 16384;
constexpr int C  = 256;
constexpr int H  = 8;
constexpr int Dh = 32;          // head dim
constexpr int TM = 64;          // tokens per block tile
constexpr int SROW = C + 4;     // 260 floats/row: 1040B -> 16B-aligned rows, bank-rotating
constexpr int CHUNK = 256;      // tokens per kv-reduction block
constexpr float EPS_ATTN = 1e-6f;
constexpr float EPS_LN   = 1e-5f;

typedef float v2f __attribute__((ext_vector_type(2)));
typedef float v8f __attribute__((ext_vector_type(8)));
typedef unsigned int v4u __attribute__((ext_vector_type(4)));
typedef int v4i __attribute__((ext_vector_type(4)));
typedef int v8i __attribute__((ext_vector_type(8)));

#if defined(__has_builtin)
#if __has_builtin(__builtin_amdgcn_tensor_load_to_lds) && \
    __has_builtin(__builtin_amdgcn_s_wait_tensorcnt)
#define USE_TDM 1
#endif
#endif

// ---------------- WMMA helpers (fp32, 16x16x4) ----------------
__device__ __forceinline__ v8f wmma4(v2f a, v2f b, v8f c) {
  // (neg_a, A, neg_b, B, c_mod, C, reuse_a, reuse_b)
  return __builtin_amdgcn_wmma_f32_16x16x4_f32(false, a, false, b, (short)0, c,
                                               false, false);
}

__device__ __forceinline__ float phi_elu1(float v) {
  return v > 0.0f ? v + 1.0f : __expf(v);
}

// LDS byte offset of a shared-memory pointer (low 32 bits of the flat address).
__device__ __forceinline__ unsigned int lds_addr32(const void* p) {
  return (unsigned int)(unsigned long long)p;
}

#ifdef USE_TDM
// Tensor Data Mover: 2D fp32 tile (tile_d1 rows x tile_d0 elems) global -> LDS,
// with an LDS row pad of `pad_dwords` DWORDs after every `tile_d0` DWORDs.
// Group0: count=1, lds_addr, 57b global addr, type=2 (image).
// Group1: data_size=4B, pad fields, tensor/tile dims, dim0 stride.
// 6-arg builtin form: (g0:v4u, g1:v8i, g2:v4i, g3:v4i, g4:v8i, cpol:i32).
__device__ __forceinline__ void tdm_load_tile_2d_f32(
    unsigned int lds_addr, const void* gptr,
    unsigned int tensor_d0, unsigned int tensor_d1,
    unsigned int tile_d0, unsigned int tile_d1,
    unsigned int row_stride_elems,
    unsigned int pad_interval_code, unsigned int pad_dwords) {
  unsigned long long ga = (unsigned long long)gptr;
  v4u g0;
  g0[0] = 1u;                                        // count=1, user mode
  g0[1] = lds_addr;                                  // lds_addr[31:0]
  g0[2] = (unsigned int)(ga & 0xffffffffu);          // global_addr[31:0]
  g0[3] = (unsigned int)((ga >> 32) & 0x01ffffffu)   // global_addr[56:32]
          | (2u << 30);                              // type=2
  unsigned int w0 = (2u << 16);                      // data_size: 4 bytes
  if (pad_dwords)
    w0 |= (1u << 20) | (pad_interval_code << 22) | ((pad_dwords - 1u) << 25);
  v8i g1;
  g1[0] = (int)w0;
  g1[1] = (int)((tensor_d0 & 0xffffu) << 16);        // barrier_addr=0 | dim0[15:0]
  g1[2] = (int)(((tensor_d0 >> 16) & 0xffffu) | ((tensor_d1 & 0xffffu) << 16));
  g1[3] = (int)(((tensor_d1 >> 16) & 0xffffu) | ((tile_d0 & 0xffffu) << 16));
  g1[4] = (int)(tile_d1 & 0xffffu);                  // tile_dim1 | tile_dim2=0
  g1[5] = (int)row_stride_elems;                     // tensor_dim0_stride[31:0]
  g1[6] = 0;                                         // stride hi | dim1_stride lo
  g1[7] = 0;
  v4i z4 = {0, 0, 0, 0};
  v8i z8 = {0, 0, 0, 0, 0, 0, 0, 0};
  __builtin_amdgcn_tensor_load_to_lds(g0, g1, z4, z4, z8, 0);
}
#endif

// Stage a [TM x C] fp32 token tile from global into LDS with SROW-padded rows.
// TDM path: one wave issues the DMA, waits TENSORcnt; manual path: all threads.
__device__ __forceinline__ void stage_tile_to_lds(
    float* S, const float* __restrict__ G, int tokbase, int tid) {
#ifdef USE_TDM
  if (__builtin_amdgcn_readfirstlane(tid >> 5) == 0) {
    tdm_load_tile_2d_f32(lds_addr32(S), G + (size_t)tokbase * C,
                         /*tensor_d0=*/C, /*tensor_d1=*/TM,
                         /*tile_d0=*/C, /*tile_d1=*/TM,
                         /*row_stride=*/C,
                         /*pad_interval 256 DWORDs=*/7u,
                         /*pad=*/(unsigned)(SROW - C));
    __builtin_amdgcn_s_wait_tensorcnt(0);
  }
#else
  for (int i = tid; i < TM * (C / 4); i += 256) {
    int row = i >> 6;
    int cv  = i & 63;
    float4 v = ((const float4*)(G + (size_t)(tokbase + row) * C))[cv];
    *(float4*)(S + row * SROW + cv * 4) = v;
  }
#endif
}

// A fragment (16 rows x 4 K) from LDS tile with row stride SROW.
// Lanes 0-15: K = kc,kc+1 ; lanes 16-31: K = kc+2,kc+3 (ISA 16x4 f32 A layout).
__device__ __forceinline__ v2f afrag_lds(const float* S, int rbase, int kc, int lane) {
  int row = rbase + (lane & 15);
  int hi  = lane >> 4;
  return *(const v2f*)(S + row * SROW + kc + 2 * hi);
}

// A fragment from global row-major [tokens, C] matrix.
__device__ __forceinline__ v2f afrag_gx(const float* X, int tokbase, int rbase,
                                        int kc, int lane) {
  int tok = tokbase + rbase + (lane & 15);
  int hi  = lane >> 4;
  return *(const v2f*)(X + (size_t)tok * C + kc + 2 * hi);
}

// B fragment (4 K x 16 cols) from global row-major weight [K, C].
// VGPR j holds rows kc+2*hi+j (ISA B-row striping).
__device__ __forceinline__ v2f bfrag_gw(const float* W, int kc, int cbase, int lane) {
  int col = cbase + (lane & 15);
  int hi  = lane >> 4;
  const float* p = W + (size_t)(kc + 2 * hi) * C + col;
  v2f b;
  b[0] = p[0];
  b[1] = p[C];
  return b;
}

// D (16x16 f32) -> LDS tile. VGPR j: rows rbase+j (lanes 0-15) / rbase+j+8 (16-31).
__device__ __forceinline__ void store_d_lds(float* S, int rbase, int cbase,
                                            int lane, v8f d) {
  int col = cbase + (lane & 15);
  int hi  = lane >> 4;
#pragma unroll
  for (int j = 0; j < 8; ++j) S[(rbase + j + 8 * hi) * SROW + col] = d[j];
}

// =================== K1: QKV projection ===================
// Per block: 64-token x-tile in LDS (TDM-staged); 3 matrices x 16 col-tiles = 48 strips.
__global__ void __launch_bounds__(256) qkv_kernel(
    const float* __restrict__ x, const float* __restrict__ Wq,
    const float* __restrict__ Wk, const float* __restrict__ Wv,
    float* __restrict__ Qb, float* __restrict__ Kb, float* __restrict__ Vb) {
  extern __shared__ float xs[];
  const int tid  = threadIdx.x;
  const int lane = tid & 31;
  const int wave = tid >> 5;
  const int tokbase = blockIdx.x * TM;

  stage_tile_to_lds(xs, x, tokbase, tid);
  __syncthreads();

  const v8f vzero = {0.f, 0.f, 0.f, 0.f, 0.f, 0.f, 0.f, 0.f};
  for (int strip = wave; strip < 48; strip += 8) {
    // strip index is wave-uniform; make it SGPR so branches stay scalar.
    const int mi = __builtin_amdgcn_readfirstlane(strip >> 4);   // 0:Q 1:K 2:V
    const int cb = (__builtin_amdgcn_readfirstlane(strip) & 15) * 16;
    const float* W = (mi == 0) ? Wq : ((mi == 1) ? Wk : Wv);
    float* dst     = (mi == 0) ? Qb : ((mi == 1) ? Kb : Vb);
    v8f acc[4] = {vzero, vzero, vzero, vzero};
#pragma unroll 4
    for (int kc = 0; kc < C; kc += 4) {
      v2f b = bfrag_gw(W, kc, cb, lane);
#pragma unroll
      for (int r = 0; r < 4; ++r)
        acc[r] = wmma4(afrag_lds(xs, r * 16, kc, lane), b, acc[r]);
    }
    const int hi  = lane >> 4;
    const int col = cb + (lane & 15);
#pragma unroll
    for (int r = 0; r < 4; ++r) {
#pragma unroll
      for (int j = 0; j < 8; ++j) {
        int tok = tokbase + r * 16 + j + 8 * hi;
        float v = acc[r][j];
        v = (mi < 2) ? phi_elu1(v) : v * (1.0f / (float)L);
        dst[(size_t)tok * C + col] = v;
      }
    }
  }
}

// =================== K2: KV / Ksum reduction ===================
// grid = (L/CHUNK, N*H); thread (d, e-quad) accumulates a 1x4 slice of KV[32][32].
__global__ void __launch_bounds__(256) kv_kernel(
    const float* __restrict__ Kb, const float* __restrict__ Vb,
    float* __restrict__ KV, float* __restrict__ Ks) {
  const int nh = blockIdx.y;
  const int t  = threadIdx.x;
  const int d  = t >> 3;
  const int e0 = (t & 7) * 4;
  const int n  = nh >> 3;
  const int h  = nh & 7;
  size_t base = ((size_t)n * L + (size_t)blockIdx.x * CHUNK) * C + h * Dh;
  float a0 = 0.f, a1 = 0.f, a2 = 0.f, a3 = 0.f, ks = 0.f;
  for (int s = 0; s < CHUNK; ++s) {
    size_t o = base + (size_t)s * C;
    float  kd = Kb[o + d];
    float4 vv = *(const float4*)(Vb + o + e0);
    a0 = fmaf(kd, vv.x, a0);
    a1 = fmaf(kd, vv.y, a1);
    a2 = fmaf(kd, vv.z, a2);
    a3 = fmaf(kd, vv.w, a3);
    ks += kd;
  }
  float* kvp = KV + ((size_t)nh * Dh + d) * Dh + e0;
  atomicAdd(kvp + 0, a0);
  atomicAdd(kvp + 1, a1);
  atomicAdd(kvp + 2, a2);
  atomicAdd(kvp + 3, a3);
  if ((t & 7) == 0) atomicAdd(Ks + nh * Dh + d, ks);
}

// =================== K3: fused attention-msg + Wm+LN1 + MLP + LN2 + residual ===================
__global__ void __launch_bounds__(256) fused_kernel(
    const float* __restrict__ x,  const float* __restrict__ Qb,
    const float* __restrict__ KV, const float* __restrict__ Ks,
    const float* __restrict__ Wm, const float* __restrict__ W1,
    const float* __restrict__ W2,
    const float* __restrict__ g1, const float* __restrict__ b1,
    const float* __restrict__ g2, const float* __restrict__ b2,
    float* __restrict__ out) {
  extern __shared__ float sm[];
  float* ms  = sm;                    // msg tile, later relu(h) tile
  float* m1s = sm + TM * SROW;        // Q tile (stage A), LN1 out, pre-LN2 accum
  float* mu  = sm + 2 * TM * SROW;    // [TM]
  float* rsd = mu + TM;               // [TM]
  const int tid  = threadIdx.x;
  const int lane = tid & 31;
  const int wave = tid >> 5;
  const int tokbase = blockIdx.x * TM;
  const int n = tokbase / L;
  const v8f vzero = {0.f, 0.f, 0.f, 0.f, 0.f, 0.f, 0.f, 0.f};

  // Stage the Q tile into m1s (free until stage B overwrites it).
  stage_tile_to_lds(m1s, Qb, tokbase, tid);
  __syncthreads();

  // ---- Stage A: msg[r][h*32+e] = L/(Q.Ksum+eps) * (Q @ KV) ----
#pragma unroll
  for (int it = 0; it < 2; ++it) {
    int idx = tid + it * 256;       // 512 (token,head) pairs
    int r = idx >> 3, h = idx & 7;
    const float* qp = m1s + r * SROW + h * Dh;   // Q from LDS
    float4 q[8];
#pragma unroll
    for (int i = 0; i < 8; ++i) q[i] = ((const float4*)qp)[i];
    const float* ksp = Ks + (n * H + h) * Dh;
    float den = EPS_ATTN;
#pragma unroll
    for (int i = 0; i < 8; ++i) {
      float4 kk = ((const float4*)ksp)[i];
      den += q[i].x * kk.x + q[i].y * kk.y + q[i].z * kk.z + q[i].w * kk.w;
    }
    float z = (float)L / den;
    float4 a[8];
#pragma unroll
    for (int e = 0; e < 8; ++e) a[e] = make_float4(0.f, 0.f, 0.f, 0.f);
    const float* kvp = KV + (size_t)(n * H + h) * Dh * Dh;
#pragma unroll 2
    for (int dv = 0; dv < 8; ++dv) {
      float4 qv = q[dv];
#pragma unroll
      for (int sub = 0; sub < 4; ++sub) {
        float qd = (sub == 0) ? qv.x : (sub == 1) ? qv.y : (sub == 2) ? qv.z : qv.w;
        const float4* row = (const float4*)(kvp + (dv * 4 + sub) * Dh);
#pragma unroll
        for (int e = 0; e < 8; ++e) {
          float4 rv = row[e];
          a[e].x = fmaf(qd, rv.x, a[e].x);
          a[e].y = fmaf(qd, rv.y, a[e].y);
          a[e].z = fmaf(qd, rv.z, a[e].z);
          a[e].w = fmaf(qd, rv.w, a[e].w);
        }
      }
    }
    float* mp = ms + r * SROW + h * Dh;
#pragma unroll
    for (int e = 0; e < 8; ++e) {
      float4 o;
      o.x = a[e].x * z; o.y = a[e].y * z; o.z = a[e].z * z; o.w = a[e].w * z;
      ((float4*)mp)[e] = o;
    }
  }
  __syncthreads();

  // ---- Stage B: m1s = msg @ Wm (WMMA) ----
  for (int strip = wave; strip < 16; strip += 8) {
    const int cb = strip * 16;
    v8f acc[4] = {vzero, vzero, vzero, vzero};
#pragma unroll 4
    for (int kc = 0; kc < C; kc += 4) {
      v2f b = bfrag_gw(Wm, kc, cb, lane);
#pragma unroll
      for (int r = 0; r < 4; ++r)
        acc[r] = wmma4(afrag_lds(ms, r * 16, kc, lane), b, acc[r]);
    }
#pragma unroll
    for (int r = 0; r < 4; ++r) store_d_lds(m1s, r * 16, cb, lane, acc[r]);
  }
  __syncthreads();

  // ---- LN1 (in place on m1s) ----
  if (tid < TM) {
    const float* row = m1s + tid * SROW;
    float s = 0.f, s2 = 0.f;
    for (int c2 = 0; c2 < C; ++c2) { float v = row[c2]; s += v; s2 = fmaf(v, v, s2); }
    float m = s * (1.0f / C);
    mu[tid]  = m;
    rsd[tid] = rsqrtf(s2 * (1.0f / C) - m * m + EPS_LN);
  }
  __syncthreads();
  for (int i = tid; i < TM * C; i += 256) {
    int r = i >> 8, c2 = i & (C - 1);
    float* p = m1s + r * SROW + c2;
    *p = (*p - mu[r]) * rsd[r] * g1[c2] + b1[c2];
  }
  __syncthreads();

  // ---- Stage C: ms = relu([x | m1] @ W1), K = 512 split into two 256 halves ----
  for (int strip = wave; strip < 16; strip += 8) {
    const int cb = strip * 16;
    v8f acc[4] = {vzero, vzero, vzero, vzero};
#pragma unroll 2
    for (int kc = 0; kc < C; kc += 4) {      // x-half: A frags straight from global (L0/L2 hits)
      v2f b = bfrag_gw(W1, kc, cb, lane);
#pragma unroll
      for (int r = 0; r < 4; ++r)
        acc[r] = wmma4(afrag_gx(x, tokbase, r * 16, kc, lane), b, acc[r]);
    }
#pragma unroll 2
    for (int kc = 0; kc < C; kc += 4) {      // msg-half from LDS
      v2f b = bfrag_gw(W1 + (size_t)C * C, kc, cb, lane);
#pragma unroll
      for (int r = 0; r < 4; ++r)
        acc[r] = wmma4(afrag_lds(m1s, r * 16, kc, lane), b, acc[r]);
    }
    const int hi = lane >> 4, col = cb + (lane & 15);
#pragma unroll
    for (int r = 0; r < 4; ++r)
#pragma unroll
      for (int j = 0; j < 8; ++j)
        ms[(r * 16 + j + 8 * hi) * SROW + col] = fmaxf(acc[r][j], 0.f);
  }
  __syncthreads();

  // ---- Stage D: m1s = h @ W2 (WMMA) ----
  for (int strip = wave; strip < 16; strip += 8) {
    const int cb = strip * 16;
    v8f acc[4] = {vzero, vzero, vzero, vzero};
#pragma unroll 4
    for (int kc = 0; kc < C; kc += 4) {
      v2f b = bfrag_gw(W2, kc, cb, lane);
#pragma unroll
      for (int r = 0; r < 4; ++r)
        acc[r] = wmma4(afrag_lds(ms, r * 16, kc, lane), b, acc[r]);
    }
#pragma unroll
    for (int r = 0; r < 4; ++r) store_d_lds(m1s, r * 16, cb, lane, acc[r]);
  }
  __syncthreads();

  // ---- LN2 + residual ----
  if (tid < TM) {
    const float* row = m1s + tid * SROW;
    float s = 0.f, s2 = 0.f;
    for (int c2 = 0; c2 < C; ++c2) { float v = row[c2]; s += v; s2 = fmaf(v, v, s2); }
    float m = s * (1.0f / C);
    mu[tid]  = m;
    rsd[tid] = rsqrtf(s2 * (1.0f / C) - m * m + EPS_LN);
  }
  __syncthreads();
  for (int i = tid; i < TM * C; i += 256) {
    int r = i >> 8, c2 = i & (C - 1);
    size_t go = (size_t)(tokbase + r) * C + c2;
    float v = (m1s[r * SROW + c2] - mu[r]) * rsd[r] * g2[c2] + b2[c2];
    out[go] = x[go] + v;
  }
}

// =================== launch ===================
extern "C" void kernel_launch(void* const* d_in, const int* in_sizes, int n_in,
                              void* d_out, int out_size, void* d_ws, size_t ws_size,
                              hipStream_t stream) {
  (void)in_sizes; (void)n_in; (void)out_size; (void)ws_size;
  const float* x  = (const float*)d_in[0];
  const float* Wq = (const float*)d_in[1];
  const float* Wk = (const float*)d_in[2];
  const float* Wv = (const float*)d_in[3];
  const float* Wm = (const float*)d_in[4];
  const float* W1 = (const float*)d_in[5];
  const float* W2 = (const float*)d_in[6];
  const float* g1 = (const float*)d_in[7];
  const float* b1 = (const float*)d_in[8];
  const float* g2 = (const float*)d_in[9];
  const float* b2 = (const float*)d_in[10];
  float* out = (float*)d_out;

  const size_t NL = (size_t)Nn * L;
  float* Qb  = (float*)d_ws;
  float* Kb  = Qb + NL * C;
  float* Vb  = Kb + NL * C;
  float* KVb = Vb + NL * C;
  float* Ksb = KVb + (size_t)Nn * H * Dh * Dh;

  (void)hipMemsetAsync(KVb, 0,
                       (size_t)(Nn * H * Dh * Dh + Nn * H * Dh) * sizeof(float),
                       stream);

  const int nblocks = (int)(NL / TM);  // 1024
  const size_t lds_qkv = (size_t)TM * SROW * sizeof(float);
  hipLaunchKernelGGL(qkv_kernel, dim3(nblocks), dim3(256), lds_qkv, stream,
                     x, Wq, Wk, Wv, Qb, Kb, Vb);
  hipLaunchKernelGGL(kv_kernel, dim3(L / CHUNK, Nn * H), dim3(256), 0, stream,
                     Kb, Vb, KVb, Ksb);
  const size_t lds_f = (size_t)(2 * TM * SROW + 2 * TM) * sizeof(float);
  hipLaunchKernelGGL(fused_kernel, dim3(nblocks), dim3(256), lds_f, stream,
                     x, Qb, KVb, Ksb, Wm, W1, W2, g1, b1, g2, b2, out);
}